// GNN_28312424415242
// MI455X (gfx1250) — compile-verified
//
#include <hip/hip_runtime.h>
#include <math.h>

typedef __attribute__((ext_vector_type(2))) float v2f;
typedef __attribute__((ext_vector_type(8))) float v8f;

#if __has_builtin(__builtin_amdgcn_wmma_f32_16x16x4_f32)
#define HAVE_WMMA_F32 1
#endif

#define WPB  4    // waves (16-col N tiles) per block
#define KC   32   // K-chunk staged in LDS
#define BSTR 80   // LDS row stride (floats) for B panel: hi/lo half-waves hit
                  // disjoint bank groups (2*80 mod 64 == 32)

// guaranteed-native f32 global atomic add (no-return, STOREcnt-tracked)
__device__ __forceinline__ void gatomic_add_f32(float* p, float v) {
    asm volatile("global_atomic_add_f32 %0, %1, off" : : "v"(p), "v"(v) : "memory");
}

// ---------------------------------------------------------------------------
// C[M,N] = A[M,K] @ B[K,N] + bias[N]   (fp32, V_WMMA_F32_16X16X4_F32)
// grid = (M/16, ceil(N/(16*WPB))), block = 32*WPB
// A tile (16xKC) and B panel (KCx64) staged in LDS; inner loop = ds + wmma.
// ---------------------------------------------------------------------------
__global__ void gat_gemm_wmma_f32(const float* __restrict__ A,
                                  const float* __restrict__ B,
                                  const float* __restrict__ bias,
                                  float* __restrict__ C,
                                  int M, int N, int K) {
    __shared__ float As[16][KC + 1];
    __shared__ float Bs[KC * BSTR];
    const int mt      = blockIdx.x;
    const int wave    = threadIdx.x >> 5;
    const int lane    = threadIdx.x & 31;
    const int nt      = blockIdx.y * WPB + wave;   // 16-wide N tile of this wave
    const int colBase = blockIdx.y * (16 * WPB);   // block's 64-col B panel
    const int l16     = lane & 15;
    const bool hi     = lane >= 16;
    const bool act    = (nt * 16) < N;

    v8f acc = {};
    for (int k0 = 0; k0 < K; k0 += KC) {
        // cooperative coalesced A tile (16 x KC)
        for (int idx = threadIdx.x; idx < 16 * KC; idx += blockDim.x) {
            int r = idx / KC, c = idx % KC;
            As[r][c] = A[(size_t)(mt * 16 + r) * K + (k0 + c)];
        }
        // cooperative coalesced B panel (KC x 64)
        for (int idx = threadIdx.x; idx < KC * 64; idx += blockDim.x) {
            int r = idx >> 6, c = idx & 63;
            int col = colBase + c;
            Bs[r * BSTR + c] = (col < N) ? B[(size_t)(k0 + r) * N + col] : 0.0f;
        }
        __syncthreads();
        if (act) {
            const int bcol = wave * 16 + l16;
#if defined(HAVE_WMMA_F32)
            for (int kk = 0; kk < KC; kk += 4) {
                // A 16x4: lanes 0-15 = rows, v0/v1 = K,K+1 (hi half K+2,K+3)
                v2f a, b;
                int ka = kk + (hi ? 2 : 0);
                a.x = As[l16][ka];
                a.y = As[l16][ka + 1];
                // B 4x16: row K striped across lanes (hi half rows K+2/K+3)
                b.x = Bs[(kk + (hi ? 2 : 0)) * BSTR + bcol];
                b.y = Bs[(kk + (hi ? 3 : 1)) * BSTR + bcol];
                acc = __builtin_amdgcn_wmma_f32_16x16x4_f32(
                    false, a, false, b, (short)0, acc, false, false);
            }
#else
            for (int kk = 0; kk < KC; ++kk) {
                float bv = Bs[kk * BSTR + bcol];
#pragma unroll
                for (int r = 0; r < 8; ++r)
                    acc[r] = fmaf(As[r + (hi ? 8 : 0)][kk], bv, acc[r]);
            }
#endif
        }
        __syncthreads();
    }
    if (act) {
        int col = nt * 16 + l16;
        float bv = bias[col];
        // D 16x16: VGPR r -> row r (lanes 0-15) / row r+8 (lanes 16-31)
#pragma unroll
        for (int r = 0; r < 8; ++r) {
            int row = mt * 16 + r + (hi ? 8 : 0);
            C[(size_t)row * N + col] = acc[r] + bv;
        }
    }
}

// ---------------------------------------------------------------------------
// monotone float <-> int key for atomicMax-based segment_max
// ---------------------------------------------------------------------------
__device__ __forceinline__ int enc_f(float f) {
    int i = __float_as_int(f);
    return (i >= 0) ? i : (i ^ 0x7fffffff);
}
__device__ __forceinline__ float dec_f(int k) {
    return __int_as_float((k >= 0) ? k : (k ^ 0x7fffffff));
}
#define ENC_NEG_INF ((int)0x807FFFFF)

// ---------------------------------------------------------------------------
// self-loop attr: deg + per-dst edge_attr sums (thread per (edge, channel))
// ---------------------------------------------------------------------------
__global__ void gat_deg_loop(const int* __restrict__ dst,
                             const float* __restrict__ ea,
                             float* __restrict__ deg,
                             float* __restrict__ loopsum, int E) {
    int t = blockIdx.x * blockDim.x + threadIdx.x;
    int e = t >> 4, c = t & 15;
    if (e >= E) return;
    int d = dst[e];
    gatomic_add_f32(&loopsum[(size_t)d * 16 + c], ea[(size_t)e * 16 + c]);
    if (c == 0) gatomic_add_f32(&deg[d], 1.0f);
}

__global__ void gat_loop_div(float* __restrict__ loopsum,
                             const float* __restrict__ deg, int N) {
    int t = blockIdx.x * blockDim.x + threadIdx.x;
    if (t >= N * 16) return;
    float dg = deg[t >> 4];
    loopsum[t] = loopsum[t] / fmaxf(dg, 1.0f);
}

// out rows <- bias (folds "+bias" into scatter accumulator); reset softmax state
__global__ void gat_init_layer(float* __restrict__ out, const float* __restrict__ bias,
                               int* __restrict__ mkey, float* __restrict__ ssum,
                               int N, int C) {
    int t = blockIdx.x * blockDim.x + threadIdx.x;
    if (t >= N * C) return;
    out[t] = bias[t % C];
    if (t % C == 0) { mkey[t / C] = ENC_NEG_INF; ssum[t / C] = 0.0f; }
}

// ---------------------------------------------------------------------------
// edge scores: e = att . leaky_relu(xl[src] + xr[dst] + ea@We)
// one wave32 per edge; ee computed on the fly (We staged in LDS)
// ---------------------------------------------------------------------------
template <int C>
__global__ void gat_escore(const int* __restrict__ src, const int* __restrict__ dst,
                           const float* __restrict__ ea, const float* __restrict__ loopat,
                           const float* __restrict__ xl, const float* __restrict__ xr,
                           const float* __restrict__ We, const float* __restrict__ att,
                           float* __restrict__ esc, int* __restrict__ mkey,
                           int E, int N) {
    constexpr int F = C / 32;
    __shared__ float sWe[16 * C];
    __shared__ float sAtt[C];
    for (int i = threadIdx.x; i < 16 * C; i += blockDim.x) sWe[i] = We[i];
    for (int i = threadIdx.x; i < C; i += blockDim.x) sAtt[i] = att[i];
    __syncthreads();

    const int lane = threadIdx.x & 31;
    const int ew   = (blockIdx.x * blockDim.x + threadIdx.x) >> 5;
    const int Etot = E + N;
    if (ew >= Etot) return;

    int s, d; const float* ap;
    if (ew < E) { s = src[ew]; d = dst[ew]; ap = ea + (size_t)ew * 16; }
    else        { s = d = ew - E;           ap = loopat + (size_t)s * 16; }

    // one coalesced 64B read of the edge attr, broadcast via shfl
    float av = (lane < 16) ? ap[lane] : 0.0f;
    float attr[16];
#pragma unroll
    for (int k = 0; k < 16; ++k) attr[k] = __shfl(av, k, 32);

    const float* xlp = xl + (size_t)s * C;
    const float* xrp = xr + (size_t)d * C;
    float part = 0.0f;
#pragma unroll
    for (int f = 0; f < F; ++f) {
        int c = lane + f * 32;
        float ee = 0.0f;
#pragma unroll
        for (int k = 0; k < 16; ++k) ee = fmaf(attr[k], sWe[k * C + c], ee);
        float z = xlp[c] + xrp[c] + ee;
        z = (z > 0.0f) ? z : 0.2f * z;          // leaky_relu(0.2)
        part = fmaf(z, sAtt[c], part);
    }
#pragma unroll
    for (int m = 16; m > 0; m >>= 1) part += __shfl_xor(part, m, 32);
    if (lane == 0) {
        esc[ew] = part;
        atomicMax(&mkey[d], enc_f(part));        // segment_max (native int atomic)
    }
}

// a = exp(e - m[dst]); s[dst] += a
__global__ void gat_expsum(const int* __restrict__ dst, const float* __restrict__ esc,
                           const int* __restrict__ mkey, float* __restrict__ aexp,
                           float* __restrict__ ssum, int E, int N) {
    int e = blockIdx.x * blockDim.x + threadIdx.x;
    if (e >= E + N) return;
    int d = (e < E) ? dst[e] : (e - E);
    float m = dec_f(mkey[d]);
    if (!isfinite(m)) m = 0.0f;
    float a = expf(esc[e] - m);
    aexp[e] = a;
    gatomic_add_f32(&ssum[d], a);
}

// out[dst] += (a/(s+eps)) * xl[src]   (one wave32 per edge)
template <int C>
__global__ void gat_scatter(const int* __restrict__ src, const int* __restrict__ dst,
                            const float* __restrict__ aexp, const float* __restrict__ ssum,
                            const float* __restrict__ xl, float* __restrict__ out,
                            int E, int N) {
    constexpr int F = C / 32;
    const int lane = threadIdx.x & 31;
    const int ew   = (blockIdx.x * blockDim.x + threadIdx.x) >> 5;
    if (ew >= E + N) return;
    int s, d;
    if (ew < E) { s = src[ew]; d = dst[ew]; } else { s = d = ew - E; }
    float alpha = aexp[ew] / (ssum[d] + 1e-16f);
    const float* xlp = xl + (size_t)s * C;
    float* op = out + (size_t)d * C;
#pragma unroll
    for (int f = 0; f < F; ++f) {
        int c = lane + f * 32;
        gatomic_add_f32(&op[c], alpha * xlp[c]);
    }
}

// column mean of [N,32] -> d_out[32] (d_out pre-zeroed)
__global__ void gat_mean(const float* __restrict__ out2, float* __restrict__ dout, int N) {
    int c = threadIdx.x & 31;
    int row = blockIdx.x * (blockDim.x >> 5) + (threadIdx.x >> 5);
    int stride = gridDim.x * (blockDim.x >> 5);
    float acc = 0.0f;
    for (int r = row; r < N; r += stride) acc += out2[(size_t)r * 32 + c];
    gatomic_add_f32(&dout[c], acc * (1.0f / (float)N));
}

// ---------------------------------------------------------------------------
extern "C" void kernel_launch(void* const* d_in, const int* in_sizes, int n_in,
                              void* d_out, int out_size, void* d_ws, size_t ws_size,
                              hipStream_t stream) {
    const float* x     = (const float*)d_in[0];
    const int*   ei    = (const int*)d_in[1];
    const float* ea    = (const float*)d_in[2];
    const float* Wl1   = (const float*)d_in[3];
    const float* bl1   = (const float*)d_in[4];
    const float* Wr1   = (const float*)d_in[5];
    const float* br1   = (const float*)d_in[6];
    const float* We1   = (const float*)d_in[7];
    const float* att1  = (const float*)d_in[8];
    const float* bias1 = (const float*)d_in[9];
    const float* Wl2   = (const float*)d_in[10];
    const float* bl2   = (const float*)d_in[11];
    const float* Wr2   = (const float*)d_in[12];
    const float* br2   = (const float*)d_in[13];
    const float* We2   = (const float*)d_in[14];
    const float* att2  = (const float*)d_in[15];
    const float* bias2 = (const float*)d_in[16];

    const int N    = in_sizes[0] / 128;   // 50000
    const int E    = in_sizes[1] / 2;     // 800000
    const int Etot = E + N;
    const int* srcp = ei;                 // edge_index row 0
    const int* dstp = ei + E;             // edge_index row 1

    // workspace carve (floats)
    float* W = (float*)d_ws;
    size_t o = 0;
    float* deg    = W + o; o += (size_t)N;
    float* loopat = W + o; o += (size_t)N * 16;
    float* xl1    = W + o; o += (size_t)N * 256;
    float* xr1    = W + o; o += (size_t)N * 256;
    float* h      = W + o; o += (size_t)N * 256;
    float* xl2    = W + o; o += (size_t)N * 32;
    float* xr2    = W + o; o += (size_t)N * 32;
    float* out2   = W + o; o += (size_t)N * 32;
    float* esc    = W + o; o += (size_t)Etot;
    float* aexp   = W + o; o += (size_t)Etot;
    int*   mkey   = (int*)(W + o); o += (size_t)N;
    float* ssum   = W + o; o += (size_t)N;

    // ---- self-loop attrs: deg + segment-mean of incoming edge_attr --------
    hipMemsetAsync(deg, 0, (size_t)(N + N * 16) * sizeof(float), stream);
    {
        long long t = (long long)E * 16;
        gat_deg_loop<<<(int)((t + 255) / 256), 256, 0, stream>>>(dstp, ea, deg, loopat, E);
        gat_loop_div<<<(N * 16 + 255) / 256, 256, 0, stream>>>(loopat, deg, N);
    }

    const int HID = 256, OUT = 32, NODE = 128;
    dim3 blkG(32 * WPB);
    const int epb = 256 / 32;  // edges per block (one wave32 per edge)

    // ---- layer 1 ----------------------------------------------------------
    {
        dim3 grdG(N / 16, (HID + 16 * WPB - 1) / (16 * WPB));
        gat_gemm_wmma_f32<<<grdG, blkG, 0, stream>>>(x, Wl1, bl1, xl1, N, HID, NODE);
        gat_gemm_wmma_f32<<<grdG, blkG, 0, stream>>>(x, Wr1, br1, xr1, N, HID, NODE);

        gat_init_layer<<<(N * HID + 255) / 256, 256, 0, stream>>>(h, bias1, mkey, ssum, N, HID);

        gat_escore<256><<<(Etot + epb - 1) / epb, 256, 0, stream>>>(
            srcp, dstp, ea, loopat, xl1, xr1, We1, att1, esc, mkey, E, N);
        gat_expsum<<<(Etot + 255) / 256, 256, 0, stream>>>(dstp, esc, mkey, aexp, ssum, E, N);
        gat_scatter<256><<<(Etot + epb - 1) / epb, 256, 0, stream>>>(
            srcp, dstp, aexp, ssum, xl1, h, E, N);
    }

    // ---- layer 2 ----------------------------------------------------------
    {
        dim3 grdG(N / 16, (OUT + 16 * WPB - 1) / (16 * WPB));
        gat_gemm_wmma_f32<<<grdG, blkG, 0, stream>>>(h, Wl2, bl2, xl2, N, OUT, HID);
        gat_gemm_wmma_f32<<<grdG, blkG, 0, stream>>>(h, Wr2, br2, xr2, N, OUT, HID);

        gat_init_layer<<<(N * OUT + 255) / 256, 256, 0, stream>>>(out2, bias2, mkey, ssum, N, OUT);

        gat_escore<32><<<(Etot + epb - 1) / epb, 256, 0, stream>>>(
            srcp, dstp, ea, loopat, xl2, xr2, We2, att2, esc, mkey, E, N);
        gat_expsum<<<(Etot + 255) / 256, 256, 0, stream>>>(dstp, esc, mkey, aexp, ssum, E, N);
        gat_scatter<32><<<(Etot + epb - 1) / epb, 256, 0, stream>>>(
            srcp, dstp, aexp, ssum, xl2, out2, E, N);
    }

    // ---- final mean -------------------------------------------------------
    hipMemsetAsync(d_out, 0, 32 * sizeof(float), stream);
    gat_mean<<<256, 256, 0, stream>>>(out2, (float*)d_out, N);
}